// NoteJitter_17867063951822
// MI455X (gfx1250) — compile-verified
//
#include <hip/hip_runtime.h>
#include <stdint.h>

// ---------------- problem constants (from reference setup_inputs) ----------------
#define B_BATCH   8
#define C_PITCH   256
#define T_FRAMES  16384
#define R_ROWS    (B_BATCH * C_PITCH)   // 2048 rows
#define THREADS   256                   // 8 wave32 per block
#define CHUNK     (T_FRAMES / THREADS)  // 64 frames per thread
#define QCAP      (T_FRAMES / 2)        // worst-case onsets per row (1010... pattern)

// ---------------- threefry2x32 (exact JAX PRNG core) ----------------
struct TF2 { uint32_t a, b; };

__host__ __device__ __forceinline__ TF2 threefry2x32(uint32_t k0, uint32_t k1,
                                                     uint32_t x0, uint32_t x1) {
  const uint32_t k2 = 0x1BD11BDAu ^ k0 ^ k1;
  x0 += k0; x1 += k1;
#define TFR(r) { x0 += x1; x1 = (x1 << (r)) | (x1 >> (32 - (r))); x1 ^= x0; }
  TFR(13) TFR(15) TFR(26) TFR(6)   x0 += k1; x1 += k2 + 1u;
  TFR(17) TFR(29) TFR(16) TFR(24)  x0 += k2; x1 += k0 + 2u;
  TFR(13) TFR(15) TFR(26) TFR(6)   x0 += k0; x1 += k1 + 3u;
  TFR(17) TFR(29) TFR(16) TFR(24)  x0 += k1; x1 += k2 + 4u;
  TFR(13) TFR(15) TFR(26) TFR(6)   x0 += k2; x1 += k0 + 5u;
#undef TFR
  TF2 r; r.a = x0; r.b = x1; return r;
}

// ---------------- kernel: one block per pitch row ----------------
__global__ __launch_bounds__(THREADS) void note_jitter_gfx1250(
    const float* __restrict__ in, float* __restrict__ out,
    uint32_t kj0, uint32_t kj1,   // k_jit  (randint key)
    uint32_t ks0, uint32_t ks1,   // k_scale (uniform key)
    uint32_t applyMask)           // bit b: apply jitter to batch item b
{
  __shared__ int      s_delta[T_FRAMES];   // 64 KB: +-1 difference array / copy staging
  __shared__ uint32_t s_queue[QCAP];       // 32 KB: packed (t | dur<<16) onset records
  __shared__ int      s_scan[THREADS];     // scan buffer (suffix-min, then prefix-sum)
  __shared__ int      s_lastact[THREADS];  // last-frame-active flag per chunk
  __shared__ int      s_qcount;

  const int r    = blockIdx.x;
  const int tid  = threadIdx.x;
  const int base = tid * CHUNK;
  const float* rowIn  = in  + (size_t)r * T_FRAMES;
  float*       rowOut = out + (size_t)r * T_FRAMES;
  const int b = r >> 8;                    // r / C_PITCH
  const bool applied = (applyMask >> b) & 1u;

  if (!applied) {
    // -------- pass-through row: async global -> LDS -> global staged copy --------
    // (CDNA5 async DMA path; bypasses VGPRs, tracked on ASYNCcnt)
    // low 32 bits of a generic LDS pointer == workgroup-relative LDS byte address
    unsigned lds = (unsigned)(size_t)(const void*)(&s_delta[0]) + (unsigned)(base * 4);
    const float* gsrc = rowIn + base;
    float*       gdst = rowOut + base;
#pragma unroll
    for (int i = 0; i < CHUNK / 4; ++i) {
      unsigned d = lds + (unsigned)(i * 16);
      const void* p = (const void*)(gsrc + i * 4);
      asm volatile("global_load_async_to_lds_b128 %0, %1, off"
                   :: "v"(d), "v"(p) : "memory");
    }
    asm volatile("s_wait_asynccnt 0" ::: "memory");
#pragma unroll
    for (int i = 0; i < CHUNK / 4; ++i) {
      unsigned d = lds + (unsigned)(i * 16);
      void* p = (void*)(gdst + i * 4);
      asm volatile("global_store_async_from_lds_b128 %0, %1, off"
                   :: "v"(p), "v"(d) : "memory");
    }
    asm volatile("s_wait_asynccnt 0" ::: "memory");
    return;
  }

  // -------- Phase 1: load chunk, build 64-bit active mask, zero delta --------
  uint64_t mask = 0;
  {
    const float4* in4 = (const float4*)(rowIn + base);
#pragma unroll
    for (int i = 0; i < CHUNK / 4; ++i) {
      float4 q = in4[i];
      uint64_t m4 = (uint64_t)(q.x > 0.f)
                  | ((uint64_t)(q.y > 0.f) << 1)
                  | ((uint64_t)(q.z > 0.f) << 2)
                  | ((uint64_t)(q.w > 0.f) << 3);
      mask |= m4 << (4 * i);
    }
  }
  {
    int4* z4 = (int4*)(&s_delta[base]);
#pragma unroll
    for (int i = 0; i < CHUNK / 4; ++i) z4[i] = make_int4(0, 0, 0, 0);
  }
  const uint64_t inact = ~mask;
  s_scan[tid]    = inact ? (base + __builtin_ctzll(inact)) : T_FRAMES; // first inactive in chunk
  s_lastact[tid] = (int)((mask >> 63) & 1u);
  if (tid == 0) s_qcount = 0;
  __syncthreads();

  // -------- Phase 2: suffix-min scan -> first inactive frame after my chunk --------
  int v = s_scan[tid];
#pragma unroll
  for (int off = 1; off < THREADS; off <<= 1) {
    int w = (tid + off < THREADS) ? s_scan[tid + off] : T_FRAMES;
    __syncthreads();
    v = min(v, w);
    s_scan[tid] = v;
    __syncthreads();
  }
  const int suffixNext = (tid + 1 < THREADS) ? s_scan[tid + 1] : T_FRAMES;
  const int prevLast   = (tid == 0) ? 0 : s_lastact[tid - 1];

  // -------- Phase 3: detect onsets + run lengths, compact into LDS queue --------
  uint64_t prevm  = (mask << 1) | (uint64_t)prevLast;    // bit f = active[f-1]
  uint64_t onsets = mask & ~prevm;
  while (onsets) {
    int f = __builtin_ctzll(onsets);
    onsets &= onsets - 1;
    int t = base + f;
    uint64_t rest = inact >> f;                          // inactive bits at/after f
    int firstIn = rest ? (t + __builtin_ctzll(rest)) : suffixNext;
    int dur = firstIn - t;                               // 1..16384, fits 16 bits
    int slot = atomicAdd(&s_qcount, 1);
    s_queue[slot] = (uint32_t)t | ((uint32_t)dur << 16);
  }
  __syncthreads();
  const int nq = s_qcount;

  // -------- Phase 4: full-occupancy threefry over compacted onsets --------
  const uint32_t rowBase = (uint32_t)r * (uint32_t)T_FRAMES;
  for (int k = tid; k < nq; k += THREADS) {
    uint32_t e = s_queue[k];
    int t   = (int)(e & 0xFFFFu);
    int dur = (int)(e >> 16);
    uint32_t i = rowBase + (uint32_t)t;                  // flat (r,t) index, < 2^25

    // randint(k_jit,(R,T),-3,4): JAX draws bits of shape (2,R,T); N=2^26, m=2^25,
    // so hi=block(i,i+2^25).a, lo=.b ; offset = ((hi%7)*(2^32%7) + lo%7) % 7
    TF2 bj = threefry2x32(kj0, kj1, i, i + 0x2000000u);
    uint32_t off7 = ((bj.a % 7u) * 4u + (bj.b % 7u)) % 7u;
    int jit = (int)off7 - 3;

    // uniform(k_scale,(R,T),0.85,1.15): bits shape (R,T); N=2^25, m=2^24
    uint32_t j = i & 0x00FFFFFFu;
    TF2 bs = threefry2x32(ks0, ks1, j, j + 0x1000000u);
    uint32_t bits = (i < 0x01000000u) ? bs.a : bs.b;
    float u = __uint_as_float((bits >> 9) | 0x3F800000u) - 1.0f;
    float scale = fmaxf(0.85f, u * (1.15f - 0.85f) + 0.85f);

    int new_on = min(T_FRAMES, max(0, t + jit));
    int nd = max(1, (int)rintf((float)dur * scale));     // jnp.round = nearest-even
    int new_off = min(T_FRAMES, new_on + nd);
    if (new_on < T_FRAMES) {                             // index T affects nothing
      atomicAdd(&s_delta[new_on], 1);
      if (new_off < T_FRAMES) atomicAdd(&s_delta[new_off], -1);
    }
  }
  __syncthreads();

  // -------- Phase 5: block prefix sum of delta --------
  int total = 0;
#pragma unroll
  for (int i = 0; i < CHUNK; ++i) total += s_delta[base + i];
  s_scan[tid] = total;
  __syncthreads();
  int vsum = total;
#pragma unroll
  for (int off = 1; off < THREADS; off <<= 1) {
    int w = (tid >= off) ? s_scan[tid - off] : 0;
    __syncthreads();
    vsum += w;
    s_scan[tid] = vsum;
    __syncthreads();
  }
  int running = vsum - total;                            // exclusive prefix for my chunk

  // -------- Phase 6: cover>0 -> 1.0, vectorized store --------
  float4* out4 = (float4*)(rowOut + base);
#pragma unroll
  for (int i = 0; i < CHUNK / 4; ++i) {
    float4 o;
    running += s_delta[base + 4 * i + 0]; o.x = (running > 0) ? 1.0f : 0.0f;
    running += s_delta[base + 4 * i + 1]; o.y = (running > 0) ? 1.0f : 0.0f;
    running += s_delta[base + 4 * i + 2]; o.z = (running > 0) ? 1.0f : 0.0f;
    running += s_delta[base + 4 * i + 3]; o.w = (running > 0) ? 1.0f : 0.0f;
    out4[i] = o;
  }
}

// ---------------- host launcher ----------------
extern "C" void kernel_launch(void* const* d_in, const int* in_sizes, int n_in,
                              void* d_out, int out_size, void* d_ws, size_t ws_size,
                              hipStream_t stream) {
  (void)in_sizes; (void)n_in; (void)d_ws; (void)ws_size; (void)out_size;
  const float* in = (const float*)d_in[0];
  float* out = (float*)d_out;

  // Reproduce JAX key derivation on host (deterministic, graph-capture safe):
  // key(1) -> raw key [0,1]; split(key,3) = threefry(key, iota(6)).reshape(3,2)
  const uint32_t k0 = 0u, k1 = 1u;
  TF2 A = threefry2x32(k0, k1, 0u, 3u);
  TF2 Bq = threefry2x32(k0, k1, 1u, 4u);
  TF2 Cq = threefry2x32(k0, k1, 2u, 5u);
  // flat order: [A.a, Bq.a, Cq.a, A.b, Bq.b, Cq.b]
  uint32_t kj0 = A.a,  kj1 = Bq.a;    // k_jit
  uint32_t ks0 = Cq.a, ks1 = A.b;     // k_scale
  uint32_t ka0 = Bq.b, ka1 = Cq.b;    // k_apply

  // apply = uniform(k_apply, (8,1,1)) < 0.5 ; bits = threefry(k_apply, iota(8))
  uint32_t bits8[8];
  for (int i = 0; i < 4; ++i) {
    TF2 t = threefry2x32(ka0, ka1, (uint32_t)i, (uint32_t)(i + 4));
    bits8[i] = t.a; bits8[i + 4] = t.b;
  }
  uint32_t applyMask = 0;
  for (int b = 0; b < B_BATCH; ++b) {
    uint32_t fb = (bits8[b] >> 9) | 0x3F800000u;
    float u; __builtin_memcpy(&u, &fb, 4); u -= 1.0f;
    if (u < 0.5f) applyMask |= (1u << b);
  }

  note_jitter_gfx1250<<<dim3(R_ROWS), dim3(THREADS), 0, stream>>>(
      in, out, kj0, kj1, ks0, ks1, applyMask);
}